// ValueNetwork_16655883174111
// MI455X (gfx1250) — compile-verified
//
#include <hip/hip_runtime.h>

typedef float v2f __attribute__((ext_vector_type(2)));
typedef float v8f __attribute__((ext_vector_type(8)));

#define CDIM 128

// ---------------------------------------------------------------- utilities

__device__ __forceinline__ void atomAddF(float* p, float v) {
#if defined(__AMDGCN__)
  unsafeAtomicAdd(p, v);   // native global_atomic_add_f32 (no CAS loop)
#else
  atomicAdd(p, v);
#endif
}

// deg[i] = 1.0 (self-loop weight)
__global__ void k_set1(float* __restrict__ p, int n) {
  int i = blockIdx.x * blockDim.x + threadIdx.x;
  if (i < n) p[i] = 1.0f;
}

// deg[col[e]] += w[e]
__global__ void k_deg_scatter(const int* __restrict__ col, const float* __restrict__ w,
                              float* __restrict__ deg, int e) {
  int i = blockIdx.x * blockDim.x + threadIdx.x;
  if (i < e) atomAddF(&deg[col[i]], w[i]);
}

// dinv[i] = rsqrt(deg[i])   (deg >= 1 always, due to self loops)
__global__ void k_rsqrt_inplace(float* __restrict__ p, int n) {
  int i = blockIdx.x * blockDim.x + threadIdx.x;
  if (i < n) p[i] = rsqrtf(p[i]);
}

// v[c] = sum_j W1[c,j] * W2[j];  v[128] = sum_j b1[j] * W2[j]
__global__ void k_fold_weights(const float* __restrict__ W1, const float* __restrict__ b1,
                               const float* __restrict__ W2, float* __restrict__ v) {
  int c = threadIdx.x;                 // one block of 128 threads
  float acc = 0.0f;
  #pragma unroll 8
  for (int j = 0; j < CDIM; ++j) acc = fmaf(W1[c * CDIM + j], W2[j], acc);
  v[c] = acc;
  if (c == 0) {
    float cc = 0.0f;
    for (int j = 0; j < CDIM; ++j) cc = fmaf(b1[j], W2[j], cc);
    v[CDIM] = cc;
  }
}

// ------------------------------------------------ t = x @ v  via f32 WMMA
// One wave per 16-row tile. K=128 as 32 chained 16x16x4 f32 WMMAs.
// A layout (ISA 16x4 f32): lane = half*16 + m holds row m, k = {2*half, 2*half+1}
// in the two A VGPRs. B is 4x16 with every column = v chunk, using the same
// (vgpr, half) -> k mapping, so the contraction is the exact fp32 dot product.
// Result column N=0: lanes 0 and 16 hold D[m,0] for m = r and m = r+8 in acc[r].
__global__ void k_matvec_wmma(const float* __restrict__ x, const float* __restrict__ v,
                              float* __restrict__ t, int ntiles) {
  int lane = threadIdx.x & 31;
  int wave = (blockIdx.x * blockDim.x + threadIdx.x) >> 5;
  if (wave >= ntiles) return;          // wave-uniform: EXEC stays all-ones
  int half = lane >> 4;                // 0 or 1
  int m    = lane & 15;                // row within tile
  const float* xr = x + (size_t)(wave * 16 + m) * CDIM;

  v8f acc = {0.0f, 0.0f, 0.0f, 0.0f, 0.0f, 0.0f, 0.0f, 0.0f};
  #pragma unroll
  for (int kk = 0; kk < CDIM; kk += 4) {
    int k0 = kk + 2 * half;
    v2f a; a.x = xr[k0]; a.y = xr[k0 + 1];
    v2f b; b.x = v[k0];  b.y = v[k0 + 1];
    acc = __builtin_amdgcn_wmma_f32_16x16x4_f32(
        /*neg_a=*/false, a, /*neg_b=*/false, b,
        /*c_mod=*/(short)0, acc, /*reuse_a=*/false, /*reuse_b=*/false);
  }
  if (m == 0) {                        // lanes 0 and 16 carry column N=0
    int out0 = wave * 16 + half * 8;
    #pragma unroll
    for (int r = 0; r < 8; ++r) t[out0 + r] = acc[r];
  }
}

// scalar tail (N not multiple of 16 — unused for N=100000 but kept general)
__global__ void k_matvec_tail(const float* __restrict__ x, const float* __restrict__ v,
                              float* __restrict__ t, int start, int n) {
  int i = start + blockIdx.x * blockDim.x + threadIdx.x;
  if (i < n) {
    const float* xr = x + (size_t)i * CDIM;
    float acc = 0.0f;
    #pragma unroll 8
    for (int j = 0; j < CDIM; ++j) acc = fmaf(xr[j], v[j], acc);
    t[i] = acc;
  }
}

// dst[i] = dinv[i]^2 * src[i] + addp[0]   (self-loop term + bias/constant)
__global__ void k_self_plus_bias(const float* __restrict__ dinv, const float* __restrict__ src,
                                 const float* __restrict__ addp, float* __restrict__ dst, int n) {
  int i = blockIdx.x * blockDim.x + threadIdx.x;
  if (i < n) {
    float d = dinv[i];
    dst[i] = d * d * src[i] + addp[0];
  }
}

// dst[col[e]] += dinv[row]*w*dinv[col] * src[row]
__global__ void k_edge_scatter(const int* __restrict__ row, const int* __restrict__ col,
                               const float* __restrict__ w, const float* __restrict__ dinv,
                               const float* __restrict__ src, float* __restrict__ dst, int e) {
  int i = blockIdx.x * blockDim.x + threadIdx.x;
  if (i < e) {
    int r = row[i], c = col[i];
    float nrm = dinv[r] * w[i] * dinv[c];
    atomAddF(&dst[c], nrm * src[r]);
  }
}

// ---------------------------------------------------------------- launcher

extern "C" void kernel_launch(void* const* d_in, const int* in_sizes, int n_in,
                              void* d_out, int out_size, void* d_ws, size_t ws_size,
                              hipStream_t stream) {
  const float* x   = (const float*)d_in[0];   // [N,128]
  const int*   ei  = (const int*)  d_in[1];   // [2,E]
  const float* w   = (const float*)d_in[2];   // [E]
  const float* W1  = (const float*)d_in[3];   // [128,128]
  const float* b1  = (const float*)d_in[4];   // [128]
  const float* W2  = (const float*)d_in[5];   // [128,1]
  const float* b2  = (const float*)d_in[6];   // [1]
  float* out = (float*)d_out;                 // [N]

  const int N = in_sizes[0] / CDIM;
  const int E = in_sizes[1] / 2;
  const int* row = ei;
  const int* col = ei + E;

  float* ws    = (float*)d_ws;
  float* dinv  = ws;               // N floats (deg, then dinv in place)
  float* t     = ws + (size_t)N;   // N floats: x @ (W1 W2)
  float* s     = ws + 2*(size_t)N; // N floats: layer-1 folded output
  float* vfold = ws + 3*(size_t)N; // 129 floats: v[0..127], c at [128]

  const int B = 256;
  auto g = [](int n, int b) { return (n + b - 1) / b; };

  // normalization
  k_set1<<<g(N, B), B, 0, stream>>>(dinv, N);
  k_deg_scatter<<<g(E, B), B, 0, stream>>>(col, w, dinv, E);
  k_rsqrt_inplace<<<g(N, B), B, 0, stream>>>(dinv, N);

  // fold both weight matrices: v = W1 @ W2, c = b1 . W2
  k_fold_weights<<<1, CDIM, 0, stream>>>(W1, b1, W2, vfold);

  // t = x @ v  (fp32 WMMA)
  int ntiles = N / 16;
  int rem    = N - ntiles * 16;
  if (ntiles > 0)
    k_matvec_wmma<<<g(ntiles * 32, B), B, 0, stream>>>(x, vfold, t, ntiles);
  if (rem > 0)
    k_matvec_tail<<<g(rem, B), B, 0, stream>>>(x, vfold, t, ntiles * 16, N);

  // layer 1 (folded): s = Agg(t) + c
  k_self_plus_bias<<<g(N, B), B, 0, stream>>>(dinv, t, vfold + CDIM, s, N);
  k_edge_scatter<<<g(E, B), B, 0, stream>>>(row, col, w, dinv, t, s, E);

  // layer 2: out = Agg(s) + b2
  k_self_plus_bias<<<g(N, B), B, 0, stream>>>(dinv, s, b2, out, N);
  k_edge_scatter<<<g(E, B), B, 0, stream>>>(row, col, w, dinv, s, out, E);
}